// RandScatter_46308337386319
// MI455X (gfx1250) — compile-verified
//
#include <hip/hip_runtime.h>

// RandScatter dispatch with constant score: routing is static (path0 = row 0,
// path1 = rows 1..3); flat-concatenated tuple output == rows 0..3 in order,
// i.e. an identity copy of 4 x 8388608 fp32 (128 MB in + 128 MB out).
// Pure HBM-bandwidth problem: floor = 256 MB / 23.3 TB/s ~= 11 us.

typedef __attribute__((ext_vector_type(4))) float floatx4;

// Exact-cover streaming copy: gridDim.x * 256 * UNROLL == n4 (host guarantees).
// 32-bit offsets -> GVS addressing (SGPR base + 32-bit VGPR offset), no
// guards, no loop: 4 NT b128 loads, one wait, 4 NT b128 stores.
__global__ __launch_bounds__(256) void
randscatter_copy_exact(const floatx4* __restrict__ src,
                       floatx4* __restrict__ dst) {
    const unsigned stride = gridDim.x * 256u;                 // float4 units
    const unsigned t = blockIdx.x * 256u + threadIdx.x;

    floatx4 v0 = __builtin_nontemporal_load(&src[t]);
    floatx4 v1 = __builtin_nontemporal_load(&src[t + stride]);
    floatx4 v2 = __builtin_nontemporal_load(&src[t + 2u * stride]);
    floatx4 v3 = __builtin_nontemporal_load(&src[t + 3u * stride]);

    __builtin_nontemporal_store(v0, &dst[t]);
    __builtin_nontemporal_store(v1, &dst[t + stride]);
    __builtin_nontemporal_store(v2, &dst[t + 2u * stride]);
    __builtin_nontemporal_store(v3, &dst[t + 3u * stride]);
}

// Generic guarded grid-stride fallback (not used for this static shape).
__global__ __launch_bounds__(256) void
randscatter_copy_generic(const floatx4* __restrict__ src,
                         floatx4* __restrict__ dst,
                         long long n4) {
    const long long nthreads = (long long)gridDim.x * blockDim.x;
    for (long long i = (long long)blockIdx.x * blockDim.x + threadIdx.x;
         i < n4; i += nthreads) {
        __builtin_nontemporal_store(__builtin_nontemporal_load(&src[i]), &dst[i]);
    }
}

// Scalar tail for element counts not divisible by 4 (not hit for this shape).
__global__ __launch_bounds__(64) void
randscatter_copy_tail(const float* __restrict__ src,
                      float* __restrict__ dst,
                      long long start, long long n) {
    long long i = start + (long long)blockIdx.x * blockDim.x + threadIdx.x;
    if (i < n)
        __builtin_nontemporal_store(__builtin_nontemporal_load(&src[i]), &dst[i]);
}

extern "C" void kernel_launch(void* const* d_in, const int* in_sizes, int n_in,
                              void* d_out, int out_size, void* d_ws, size_t ws_size,
                              hipStream_t stream) {
    (void)n_in; (void)in_sizes; (void)d_ws; (void)ws_size;

    const float* src = (const float*)d_in[0];
    float* dst = (float*)d_out;

    const long long n  = (long long)out_size;   // 33,554,432 floats
    const long long n4 = n >> 2;                // 8,388,608 float4s
    const long long tail = n - (n4 << 2);

    constexpr int BLOCK = 256;
    constexpr int UNROLL = 4;
    const long long chunk = (long long)BLOCK * UNROLL;   // float4s per block

    if (n4 > 0) {
        if ((n4 % chunk) == 0 && (n4 / chunk) <= 0x7FFFFFFFLL) {
            // Static shape path: 8192 blocks, each thread moves 4 float4s.
            int blocks = (int)(n4 / chunk);
            randscatter_copy_exact<<<blocks, BLOCK, 0, stream>>>(
                (const floatx4*)src, (floatx4*)dst);
        } else {
            long long want = (n4 + chunk - 1) / chunk;
            int blocks = (int)(want > 131072 ? 131072 : want);
            randscatter_copy_generic<<<blocks, BLOCK, 0, stream>>>(
                (const floatx4*)src, (floatx4*)dst, n4);
        }
    }
    if (tail > 0) {
        randscatter_copy_tail<<<1, 64, 0, stream>>>(src, dst, n4 << 2, n);
    }
}